// PerturbEmbedding_25821343383947
// MI455X (gfx1250) — compile-verified
//
#include <hip/hip_runtime.h>
#include <hip/hip_bf16.h>
#include <math.h>

// Problem constants (fixed by the reference harness)
#define BB      8
#define NN      20000
#define EE      320000
#define CC      128
#define DD      64
#define HOPS_N  4
#define HID     512
#define OUT_DIM 256
#define K1      (HOPS_N * DD)   // 256
#define ROWS_WG 32              // rows (M) per workgroup in the fused MLP

typedef __attribute__((ext_vector_type(2))) float v2f;
typedef __attribute__((ext_vector_type(8))) float v8f;
typedef int v4i __attribute__((vector_size(16)));   // 4 x i32, matches builtin proto

#define AS1 __attribute__((address_space(1)))
#define AS3 __attribute__((address_space(3)))

#if defined(__has_builtin)
#if __has_builtin(__builtin_amdgcn_global_load_async_to_lds_b128)
#define USE_ASYNC_STAGE 1
#endif
#endif

// ---------------------------------------------------------------------------
// 1) idx[n] = argmax_c P[c*N + n]   (first max wins, like jnp.argmax)
// ---------------------------------------------------------------------------
__global__ void pe_argmax_kernel(const float* __restrict__ P, int* __restrict__ idx) {
    int n = blockIdx.x * blockDim.x + threadIdx.x;
    if (n >= NN) return;
    float best = P[n];
    int   bi   = 0;
    for (int c = 1; c < CC; ++c) {
        float v = P[(size_t)c * NN + n];   // coalesced across threads
        if (v > best) { best = v; bi = c; }
    }
    idx[n] = bi;
}

// ---------------------------------------------------------------------------
// 2) init_emb[n, d] = emb_table[idx[n], d]
// ---------------------------------------------------------------------------
__global__ void pe_gather_kernel(const int* __restrict__ idx,
                                 const float* __restrict__ emb,
                                 float* __restrict__ init_emb) {
    int t = blockIdx.x * blockDim.x + threadIdx.x;   // grid == N*D exactly
    int n = t >> 6;
    int d = t & (DD - 1);
    init_emb[t] = emb[idx[n] * DD + d];
}

// ---------------------------------------------------------------------------
// 3) zero-fill (vectorized) for the per-hop accumulation buffer
// ---------------------------------------------------------------------------
__global__ void pe_zero4_kernel(float4* __restrict__ p) {
    size_t t = (size_t)blockIdx.x * blockDim.x + threadIdx.x;
    p[t] = make_float4(0.f, 0.f, 0.f, 0.f);
}

// ---------------------------------------------------------------------------
// 4) SpMM scatter: Hout[b, row[e], d] += Hin[b or bcast, col[e], d]
//    64 consecutive lanes per edge -> coalesced 256B gather + atomic burst.
//    H buffers are ~5MB/graph -> L2-resident atomics.
// ---------------------------------------------------------------------------
__global__ void pe_spmm_kernel(const int* __restrict__ edge_index,
                               const float* __restrict__ Hin, long hin_graph_stride,
                               float* __restrict__ Hout) {
    const int b = blockIdx.y;
    long t = (long)blockIdx.x * blockDim.x + threadIdx.x;   // grid.x == E*64/256 exactly
    int  e = (int)(t >> 6);
    int  d = (int)(t & (DD - 1));
    const int* rows = edge_index + (size_t)b * 2 * EE;      // ei[b,0,:]
    const int* cols = rows + EE;                            // ei[b,1,:]
    int r = rows[e];
    int c = cols[e];
    float v = Hin[(size_t)b * hin_graph_stride + (size_t)c * DD + d];
    atomicAdd(&Hout[((size_t)b * NN + r) * DD + d], v);
}

// ---------------------------------------------------------------------------
// 5) omega[b, n, d*HOPS + h] = H[b, n, d]   (stack(axis=-1).reshape layout)
// ---------------------------------------------------------------------------
__global__ void pe_hop_to_omega_kernel(const float* __restrict__ H,
                                       float* __restrict__ omega, int hop) {
    const int b = blockIdx.y;
    int t = blockIdx.x * blockDim.x + threadIdx.x;  // grid.x == N*D/256 exactly
    int n = t >> 6;
    int d = t & (DD - 1);
    omega[((size_t)b * NN + n) * K1 + d * HOPS_N + hop] =
        H[((size_t)b * NN + n) * DD + d];
}

// ---------------------------------------------------------------------------
// 6) Fused  x = omega @ W1 + b1 ; LayerNorm ; GELU(exact) ; out = x @ W2 + b2
//    One WG (256 thr = 8 waves) per 32 rows (two 16-row M-tiles).
//    Each wave owns a 64-column slab of HID (GEMM1) / 32 columns of OUT
//    (GEMM2) for BOTH M-tiles, so every B-fragment load feeds two WMMAs ->
//    W1/W2 are read from L2 only once per 32 rows (5 GB total vs 10 GB).
//
//    VGPR layouts per CDNA5 ISA (32-bit operands, V_WMMA_F32_16X16X4_F32):
//      A 16x4 : lanes 0-15 -> M=lane,  {v0,v1} = K{0,1}; lanes16-31 -> K{2,3}
//      B 4x16 : lanes 0-15 -> N=lane,  {v0,v1} = K{0,1}; lanes16-31 -> K{2,3}
//      C/D    : v in 0..7 -> M = v + 8*(lane>=16), N = lane&15
// ---------------------------------------------------------------------------
#define OM_LD 260   // 256 + pad: stride % 64 banks == 4 -> conflict-free col reads
#define X_LD  516   // 512 + pad

__global__ __launch_bounds__(256) void pe_mlp_kernel(
        const float* __restrict__ omega,
        const float* __restrict__ W1, const float* __restrict__ b1,
        const float* __restrict__ gamma, const float* __restrict__ beta,
        const float* __restrict__ W2, const float* __restrict__ b2,
        float* __restrict__ out) {
    __shared__ __align__(16) float s_om[ROWS_WG * OM_LD];   // 32x256 omega tile
    __shared__ __align__(16) float s_x [ROWS_WG * X_LD];    // 32x512 hidden tile
    __shared__ float s_ps[ROWS_WG * 8];
    __shared__ float s_pq[ROWS_WG * 8];
    __shared__ float s_mu[ROWS_WG];
    __shared__ float s_rs[ROWS_WG];

    const int b    = blockIdx.y;
    const int r0   = blockIdx.x * ROWS_WG;
    const int tid  = threadIdx.x;
    const int wave = tid >> 5;                // 0..7 (wave32)
    const int lane = tid & 31;
    const int mrow  = lane & 15;              // tile row (A) / tile col (B,D)
    const int khalf = (lane >> 4) << 1;       // 0 or 2
    const int half  = lane >> 4;              // D-matrix row half

    // ---- stage omega tile (32 x 256 = 32KB), async DMA to LDS if available
    const float* om = omega + ((size_t)b * NN + r0) * K1;
#ifdef USE_ASYNC_STAGE
    #pragma unroll
    for (int i = 0; i < 8; ++i) {
        const int j   = tid + i * 256;        // 2048 x b128 chunks
        const int row = j >> 6;
        const int q   = j & 63;               // float4 index within row
        const float* gp = om + (size_t)row * K1 + q * 4;
        float*       lp = &s_om[row * OM_LD + q * 4];
        __builtin_amdgcn_global_load_async_to_lds_b128(
            (AS1 v4i*)gp, (AS3 v4i*)lp, 0, 0);
    }
#if defined(__has_builtin) && __has_builtin(__builtin_amdgcn_s_wait_asynccnt)
    __builtin_amdgcn_s_wait_asynccnt(0);
#else
    asm volatile("s_wait_asynccnt 0x0" ::: "memory");
#endif
#else
    #pragma unroll
    for (int i = 0; i < 8; ++i) {
        const int j   = tid + i * 256;
        const int row = j >> 6;
        const int q   = j & 63;
        const float4 v = reinterpret_cast<const float4*>(om + (size_t)row * K1)[q];
        *reinterpret_cast<float4*>(&s_om[row * OM_LD + q * 4]) = v;
    }
#endif
    __syncthreads();

    // ---- GEMM1: wave handles HID columns [wave*64, wave*64+64), 2 M-tiles --
    v8f acc[2][4] = {};
    const int c0 = wave * 64;
    for (int kk = 0; kk < K1; kk += 4) {
        v2f A0, A1;
        A0.x = s_om[ mrow       * OM_LD + kk + khalf];
        A0.y = s_om[ mrow       * OM_LD + kk + khalf + 1];
        A1.x = s_om[(mrow + 16) * OM_LD + kk + khalf];
        A1.y = s_om[(mrow + 16) * OM_LD + kk + khalf + 1];
        #pragma unroll
        for (int t = 0; t < 4; ++t) {
            const int ct = c0 + t * 16;
            v2f Bf;
            Bf.x = W1[(size_t)(kk + khalf)     * HID + ct + mrow];
            Bf.y = W1[(size_t)(kk + khalf + 1) * HID + ct + mrow];
            acc[0][t] = __builtin_amdgcn_wmma_f32_16x16x4_f32(
                            false, A0, false, Bf, (short)0, acc[0][t], false, false);
            acc[1][t] = __builtin_amdgcn_wmma_f32_16x16x4_f32(
                            false, A1, false, Bf, (short)0, acc[1][t], false, false);
        }
    }
    // x tile (+bias) -> LDS
    #pragma unroll
    for (int mt = 0; mt < 2; ++mt) {
        #pragma unroll
        for (int t = 0; t < 4; ++t) {
            const int ct = c0 + t * 16;
            const float bv = b1[ct + mrow];
            #pragma unroll
            for (int v = 0; v < 8; ++v) {
                const int m = mt * 16 + v + half * 8;
                s_x[m * X_LD + ct + mrow] = acc[mt][t][v] + bv;
            }
        }
    }
    __syncthreads();

    // ---- LayerNorm(512) + exact GELU per row (8 threads/row, lane-strided) -
    {
        const int row = tid >> 3;             // 0..31
        const int sub = tid & 7;              // 8 threads/row, 64 elems each
        float s = 0.f, q = 0.f;
        #pragma unroll 8
        for (int j = 0; j < 64; ++j) {
            float v = s_x[row * X_LD + sub + j * 8];
            s += v; q += v * v;
        }
        s_ps[row * 8 + sub] = s;
        s_pq[row * 8 + sub] = q;
        __syncthreads();
        if (sub == 0) {
            float ss = 0.f, qq = 0.f;
            #pragma unroll
            for (int j = 0; j < 8; ++j) { ss += s_ps[row * 8 + j]; qq += s_pq[row * 8 + j]; }
            const float mu  = ss * (1.0f / HID);
            const float var = qq * (1.0f / HID) - mu * mu;
            s_mu[row] = mu;
            s_rs[row] = rsqrtf(var + 1e-5f);
        }
        __syncthreads();
        const float mu = s_mu[row], rs = s_rs[row];
        #pragma unroll 8
        for (int j = 0; j < 64; ++j) {
            const int c = sub + j * 8;
            float v = s_x[row * X_LD + c];
            v = (v - mu) * rs * gamma[c] + beta[c];
            v = 0.5f * v * (1.0f + erff(v * 0.70710678118654752f));  // exact GELU
            s_x[row * X_LD + c] = v;
        }
    }
    __syncthreads();

    // ---- GEMM2: wave handles OUT columns [wave*32, wave*32+32), 2 M-tiles --
    v8f acc2[2][2] = {};
    const int o0 = wave * 32;
    for (int kk = 0; kk < HID; kk += 4) {
        v2f A0, A1;
        A0.x = s_x[ mrow       * X_LD + kk + khalf];
        A0.y = s_x[ mrow       * X_LD + kk + khalf + 1];
        A1.x = s_x[(mrow + 16) * X_LD + kk + khalf];
        A1.y = s_x[(mrow + 16) * X_LD + kk + khalf + 1];
        #pragma unroll
        for (int t = 0; t < 2; ++t) {
            const int ct = o0 + t * 16;
            v2f Bf;
            Bf.x = W2[(size_t)(kk + khalf)     * OUT_DIM + ct + mrow];
            Bf.y = W2[(size_t)(kk + khalf + 1) * OUT_DIM + ct + mrow];
            acc2[0][t] = __builtin_amdgcn_wmma_f32_16x16x4_f32(
                             false, A0, false, Bf, (short)0, acc2[0][t], false, false);
            acc2[1][t] = __builtin_amdgcn_wmma_f32_16x16x4_f32(
                             false, A1, false, Bf, (short)0, acc2[1][t], false, false);
        }
    }
    // +bias, store
    float* op = out + ((size_t)b * NN + r0) * OUT_DIM;
    #pragma unroll
    for (int mt = 0; mt < 2; ++mt) {
        #pragma unroll
        for (int t = 0; t < 2; ++t) {
            const int ct = o0 + t * 16;
            const float bv = b2[ct + mrow];
            #pragma unroll
            for (int v = 0; v < 8; ++v) {
                const int m = mt * 16 + v + half * 8;
                op[(size_t)m * OUT_DIM + ct + mrow] = acc2[mt][t][v] + bv;
            }
        }
    }
}

// ---------------------------------------------------------------------------
// Host orchestration
// ---------------------------------------------------------------------------
static inline size_t align256(size_t x) { return (x + 255) & ~(size_t)255; }

extern "C" void kernel_launch(void* const* d_in, const int* in_sizes, int n_in,
                              void* d_out, int out_size, void* d_ws, size_t ws_size,
                              hipStream_t stream) {
    const int*   edge_index = (const int*)  d_in[0];
    /* d_in[1] = num_nodes scalar (compile-time constant NN) */
    const float* P     = (const float*)d_in[2];   // (C, N)
    const float* emb   = (const float*)d_in[3];   // (C, D)
    const float* W1    = (const float*)d_in[4];   // (256, 512)
    const float* b1    = (const float*)d_in[5];
    const float* gamma = (const float*)d_in[6];
    const float* beta  = (const float*)d_in[7];
    const float* W2    = (const float*)d_in[8];   // (512, 256)
    const float* b2    = (const float*)d_in[9];
    float* out = (float*)d_out;

    // Workspace carve-up (~252 MB): idx | init_emb | H0 | H1 | omega
    char* ws = (char*)d_ws;
    int*   idx      = (int*)ws;    ws += align256((size_t)NN * sizeof(int));
    float* init_emb = (float*)ws;  ws += align256((size_t)NN * DD * sizeof(float));
    float* H0       = (float*)ws;  ws += align256((size_t)BB * NN * DD * sizeof(float));
    float* H1       = (float*)ws;  ws += align256((size_t)BB * NN * DD * sizeof(float));
    float* omega    = (float*)ws;  // (B, N, 256)

    pe_argmax_kernel<<<(NN + 255) / 256, 256, 0, stream>>>(P, idx);
    pe_gather_kernel<<<(NN * DD) / 256, 256, 0, stream>>>(idx, emb, init_emb);

    const float* Hin = init_emb;     // hop 0 broadcasts init_emb across graphs
    long hin_stride  = 0;
    float* Hbuf[2] = {H0, H1};
    for (int h = 0; h < HOPS_N; ++h) {
        float* Hout = Hbuf[h & 1];
        pe_zero4_kernel<<<(BB * NN * DD / 4) / 256, 256, 0, stream>>>((float4*)Hout);
        dim3 gs((EE * DD) / 256, BB);
        pe_spmm_kernel<<<gs, 256, 0, stream>>>(edge_index, Hin, hin_stride, Hout);
        dim3 gc((NN * DD) / 256, BB);
        pe_hop_to_omega_kernel<<<gc, 256, 0, stream>>>(Hout, omega, h);
        Hin = Hout;
        hin_stride = (long)NN * DD;
    }

    dim3 gm(NN / ROWS_WG, BB);   // 625 x 8 workgroups, 256 threads each
    pe_mlp_kernel<<<gm, 256, 0, stream>>>(omega, W1, b1, gamma, beta, W2, b2, out);
}